// CrossAttention_DenseAVInteractions_82437602279641
// MI455X (gfx1250) — compile-verified
//
#include <hip/hip_runtime.h>
#include <math.h>

// ---------------- problem constants ----------------
#define BATCH   16
#define DCH     96
#define HW      32
#define LSEQ    1024            // 32*32
#define SEQ     (BATCH * LSEQ)  // 16384 rows
#define NST     8
#define OUTC    256
#define UPW     128
#define CHUNK   16
#define NCH     (LSEQ / CHUNK)  // 64 chunks

typedef __attribute__((ext_vector_type(2))) float v2f;
typedef __attribute__((ext_vector_type(4))) float v4f;
typedef __attribute__((ext_vector_type(8))) float v8f;

// ---------------- LayerNorm over channels (D=96), wave32 per row ----------------
__global__ void ln_kernel(const float* __restrict__ x, const float* __restrict__ g,
                          const float* __restrict__ bb, float* __restrict__ xn) {
    int gid  = blockIdx.x * blockDim.x + threadIdx.x;
    int row  = gid >> 5;            // (b*L + t)
    int lane = gid & 31;
    int b = row >> 10, t = row & 1023;
    const float* xb = x + (size_t)b * DCH * LSEQ + t;   // x[b, c, t], stride L over c
    float v0 = xb[(size_t)lane * LSEQ];
    float v1 = xb[(size_t)(lane + 32) * LSEQ];
    float v2 = xb[(size_t)(lane + 64) * LSEQ];
    float s  = v0 + v1 + v2;
    float sq = v0 * v0 + v1 * v1 + v2 * v2;
    #pragma unroll
    for (int m = 16; m > 0; m >>= 1) {
        s  += __shfl_xor(s,  m, 32);
        sq += __shfl_xor(sq, m, 32);
    }
    float mean = s * (1.f / 96.f);
    float var  = sq * (1.f / 96.f) - mean * mean;
    float inv  = rsqrtf(var + 1e-5f);
    float* orow = xn + (size_t)row * DCH;
    orow[lane]      = (v0 - mean) * inv * g[lane]      + bb[lane];
    orow[lane + 32] = (v1 - mean) * inv * g[lane + 32] + bb[lane + 32];
    orow[lane + 64] = (v2 - mean) * inv * g[lane + 64] + bb[lane + 64];
}

// ---------------- WMMA GEMM: each wave computes one 16-row strip x (NTILES*16) cols ----
// B staged in LDS once per block, K-pair interleaved: Bs2[(k/2)*N + n] = {B[k][n], B[k+1][n]}
// so each B fragment is one aligned ds_load_b64.  B element (k,n) = Bw[k*ldbk + n*ldbn].
// ACT 1 = softplus.  RESMODE 1 = add BCHW residual res[(b*96+n)*1024 + t], row=b*1024+t.
template <int NTILES, int ACT, int RESMODE>
__global__ void gemm_wmma_f32(const float* __restrict__ A, int lda,
                              const float* __restrict__ Bw, int ldbk, int ldbn,
                              const float* __restrict__ Res,
                              float* __restrict__ C, int ldc,
                              int M, int K) {
    constexpr int N = NTILES * 16;
    extern __shared__ v2f Bs2[];             // (K/2) * N entries, 8B each
    for (int i = threadIdx.x; i < (K >> 1) * N; i += blockDim.x) {
        int kp = i / N, n = i - kp * N;
        v2f t;
        t.x = Bw[(size_t)(2 * kp)     * ldbk + (size_t)n * ldbn];
        t.y = Bw[(size_t)(2 * kp + 1) * ldbk + (size_t)n * ldbn];
        Bs2[i] = t;
    }
    __syncthreads();

    int wave = (blockIdx.x * blockDim.x + threadIdx.x) >> 5;
    if (wave >= (M >> 4)) return;            // whole-wave exit: EXEC all-1 below
    int lane = threadIdx.x & 31;
    int rowBase = wave << 4;
    int r    = lane & 15;                    // M index (A) / N index (B,D)
    int half = lane >> 4;                    // lanes 16-31 hold K+2,K+3
    int kb   = half << 1;
    v8f acc[NTILES];
    #pragma unroll
    for (int j = 0; j < NTILES; ++j) acc[j] = (v8f){0.f,0.f,0.f,0.f,0.f,0.f,0.f,0.f};

    const float* arow  = A + (size_t)(rowBase + r) * lda + kb;
    const v2f*   bbase = Bs2 + half * N + r;   // fragment row for this lane's K-pair
    for (int k0 = 0; k0 < K; k0 += 4) {
        v2f a = *reinterpret_cast<const v2f*>(arow + k0);   // 8B aligned (lda even, k even)
        const v2f* bk = bbase + (k0 >> 1) * N;
        #pragma unroll
        for (int j = 0; j < NTILES; ++j) {
            v2f b = bk[j * 16];                              // single ds_load_b64
            acc[j] = __builtin_amdgcn_wmma_f32_16x16x4_f32(false, a, false, b,
                                                           (short)0, acc[j], false, false);
        }
    }
    #pragma unroll
    for (int j = 0; j < NTILES; ++j) {
        int col = j * 16 + r;
        #pragma unroll
        for (int rr = 0; rr < 8; ++rr) {
            int row = rowBase + rr + (half << 3);
            float v = acc[j][rr];
            if (ACT == 1) v = fmaxf(v, 0.f) + log1pf(__expf(-fabsf(v)));  // stable softplus
            if (RESMODE == 1) {
                int bi = row >> 10, t = row & 1023;
                v += Res[((size_t)bi * DCH + col) * LSEQ + t];
            }
            C[(size_t)row * ldc + col] = v;
        }
    }
}

// ---------------- tiny N=8 projections (B/C of SSM) ----------------
__global__ void proj8_kernel(const float* __restrict__ xn, const float* __restrict__ W,
                             float* __restrict__ out) {
    int idx = blockIdx.x * blockDim.x + threadIdx.x;     // SEQ*8
    if (idx >= SEQ * NST) return;
    int n = idx & 7;
    int row = idx >> 3;
    const float* xr = xn + (size_t)row * DCH;
    float s = 0.f;
    #pragma unroll 8
    for (int k = 0; k < DCH; ++k) s += xr[k] * W[k * NST + n];
    out[idx] = s;
}

// ---------------- chunk-parallel SSM scan ----------------
// pass1: local scan (h0=0) per (b, chunk, d) -> end state E and chunk decay P = exp(A*sum dt)
__global__ void scan_pass1(const float* __restrict__ dtb, const float* __restrict__ ub,
                           const float* __restrict__ Bpb, const float* __restrict__ Alog,
                           float* __restrict__ Ebuf, float* __restrict__ Pbuf) {
    int bc = blockIdx.x;                 // b*NCH + c
    int b = bc >> 6, c = bc & (NCH - 1);
    int d = threadIdx.x;                 // 0..95
    float A[NST], h[NST];
    #pragma unroll
    for (int n = 0; n < NST; ++n) { A[n] = -__expf(Alog[d * NST + n]); h[n] = 0.f; }
    float dtsum = 0.f;
    size_t row0 = (size_t)b * LSEQ + c * CHUNK;
    for (int t = 0; t < CHUNK; ++t) {
        size_t row = row0 + t;
        float dt = dtb[row * DCH + d];
        float u  = ub[row * DCH + d];
        float du = dt * u;
        dtsum += dt;
        #pragma unroll
        for (int n = 0; n < NST; ++n)
            h[n] = h[n] * __expf(dt * A[n]) + du * Bpb[row * NST + n];
    }
    size_t o = ((size_t)bc * DCH + d) * NST;
    #pragma unroll
    for (int n = 0; n < NST; ++n) { Ebuf[o + n] = h[n]; Pbuf[o + n] = __expf(dtsum * A[n]); }
}

// pass2: prefix over 64 chunk summaries per (b,d,n) -> chunk start states S
__global__ void scan_pass2(const float* __restrict__ Ebuf, const float* __restrict__ Pbuf,
                           float* __restrict__ Sbuf) {
    int idx = blockIdx.x * blockDim.x + threadIdx.x;     // B*DCH*NST = 12288
    if (idx >= BATCH * DCH * NST) return;
    int n = idx & 7;
    int d = (idx >> 3) % DCH;
    int b = idx / (DCH * NST);
    float S = 0.f;
    for (int c = 0; c < NCH; ++c) {
        size_t o = (((size_t)(b * NCH + c)) * DCH + d) * NST + n;
        Sbuf[o] = S;
        S = Pbuf[o] * S + Ebuf[o];
    }
}

// pass3: replay each chunk from its true start state, emit y = o + Dskip*u
__global__ void scan_pass3(const float* __restrict__ dtb, const float* __restrict__ ub,
                           const float* __restrict__ Bpb, const float* __restrict__ Cpb,
                           const float* __restrict__ Alog, const float* __restrict__ Dskip,
                           const float* __restrict__ Sbuf, float* __restrict__ yb) {
    int bc = blockIdx.x;
    int b = bc >> 6, c = bc & (NCH - 1);
    int d = threadIdx.x;
    float A[NST], h[NST];
    size_t so = ((size_t)bc * DCH + d) * NST;
    #pragma unroll
    for (int n = 0; n < NST; ++n) { A[n] = -__expf(Alog[d * NST + n]); h[n] = Sbuf[so + n]; }
    float Ds = Dskip[d];
    size_t row0 = (size_t)b * LSEQ + c * CHUNK;
    for (int t = 0; t < CHUNK; ++t) {
        size_t row = row0 + t;
        float dt = dtb[row * DCH + d];
        float u  = ub[row * DCH + d];
        float du = dt * u;
        float o = 0.f;
        #pragma unroll
        for (int n = 0; n < NST; ++n) {
            h[n] = h[n] * __expf(dt * A[n]) + du * Bpb[row * NST + n];
            o += h[n] * Cpb[row * NST + n];
        }
        yb[row * DCH + d] = o + Ds * u;
    }
}

// ---------------- fused 4x4 avg (v) + max (a) pool -> (b, p, c) layout ----------------
__global__ void pool_kernel(const float* __restrict__ ov, const float* __restrict__ oa,
                            float* __restrict__ pv, float* __restrict__ pa) {
    int idx = blockIdx.x * blockDim.x + threadIdx.x;     // B*64*96
    if (idx >= BATCH * 64 * DCH) return;
    int c = idx % DCH;
    int p = (idx / DCH) & 63;
    int b = idx / (DCH * 64);
    int i = p >> 3, j = p & 7;
    float s = 0.f, mx = -INFINITY;
    #pragma unroll
    for (int di = 0; di < 4; ++di)
        #pragma unroll
        for (int dj = 0; dj < 4; ++dj) {
            int t = (i * 4 + di) * HW + (j * 4 + dj);
            size_t off = ((size_t)b * LSEQ + t) * DCH + c;
            s += ov[off];
            mx = fmaxf(mx, oa[off]);
        }
    pv[idx] = s * (1.f / 16.f);
    pa[idx] = mx;
}

// ---------------- separable upsample + outer-sum: out[b,o,y,x] = Yv[y]+Ya[x]+bias ------
__global__ void final_kernel(const float* __restrict__ yv, const float* __restrict__ ya,
                             const float* __restrict__ bias, float* __restrict__ out) {
    __shared__ float Yv[UPW];
    __shared__ float Ya[UPW];
    int bo = blockIdx.x;           // b*256 + o
    int b  = bo >> 8;
    int o  = bo & 255;
    int tid = threadIdx.x;
    {   // 1-D half-pixel linear upsample 64 -> 128 (weights 0.75/0.25, edges clamped)
        int idx = tid & 127;
        int m = idx >> 1;
        int p0, p1; float w0, w1;
        if (idx & 1) { p0 = m; p1 = (m + 1 < 63) ? m + 1 : 63; w0 = 0.75f; w1 = 0.25f; }
        else         { p0 = (m - 1 > 0) ? m - 1 : 0; p1 = m;   w0 = 0.25f; w1 = 0.75f; }
        const float* src = (tid < 128) ? yv : ya;
        float v = w0 * src[((size_t)b * 64 + p0) * OUTC + o]
                + w1 * src[((size_t)b * 64 + p1) * OUTC + o];
        if (tid < 128) Yv[idx] = v; else Ya[idx] = v;
    }
    __syncthreads();
    float bb = bias[o];
    float* obase = out + (size_t)bo * UPW * UPW;
    #pragma unroll
    for (int it = 0; it < 16; ++it) {
        int idx4 = it * 256 + tid;             // 4096 float4 per (b,o) plane
        int y  = idx4 >> 5;
        int x4 = (idx4 & 31) << 2;
        float base = Yv[y] + bb;
        v4f val = { base + Ya[x4], base + Ya[x4 + 1], base + Ya[x4 + 2], base + Ya[x4 + 3] };
        // streamed, never re-read: non-temporal 128-bit store (TH=NT)
        __builtin_nontemporal_store(val, reinterpret_cast<v4f*>(obase + (size_t)y * UPW + x4));
    }
}

// ---------------- host orchestration ----------------
extern "C" void kernel_launch(void* const* d_in, const int* in_sizes, int n_in,
                              void* d_out, int out_size, void* d_ws, size_t ws_size,
                              hipStream_t stream) {
    const float* xv     = (const float*)d_in[0];
    const float* xa     = (const float*)d_in[1];
    const float* ln_v_g = (const float*)d_in[2];
    const float* ln_v_b = (const float*)d_in[3];
    const float* ln_a_g = (const float*)d_in[4];
    const float* ln_a_b = (const float*)d_in[5];
    const float* Win_v  = (const float*)d_in[6];
    const float* Wdt_v  = (const float*)d_in[7];
    const float* WB_v   = (const float*)d_in[8];
    const float* WC_v   = (const float*)d_in[9];
    const float* Alog_v = (const float*)d_in[10];
    const float* Dskip_v= (const float*)d_in[11];
    const float* Wout_v = (const float*)d_in[12];
    const float* Win_a  = (const float*)d_in[13];
    const float* Wdt_a  = (const float*)d_in[14];
    const float* WB_a   = (const float*)d_in[15];
    const float* WC_a   = (const float*)d_in[16];
    const float* Alog_a = (const float*)d_in[17];
    const float* Dskip_a= (const float*)d_in[18];
    const float* Wout_a = (const float*)d_in[19];
    const float* conv_w = (const float*)d_in[20];
    const float* conv_b = (const float*)d_in[21];
    float* out = (float*)d_out;

    float* ws = (float*)d_ws;
    const size_t bufA = (size_t)SEQ * DCH;               // 1,572,864 floats
    float* xvn  = ws + 0 * bufA;                         // reused for y_v
    float* xan  = ws + 1 * bufA;                         // reused for y_a
    float* u_v  = ws + 2 * bufA;
    float* dt_v = ws + 3 * bufA;                         // reused for out_v
    float* u_a  = ws + 4 * bufA;
    float* dt_a = ws + 5 * bufA;                         // reused for out_a
    float* small0 = ws + 6 * bufA;
    const size_t bufS = (size_t)SEQ * NST;               // 131,072 floats
    float* Bp_v = small0 + 0 * bufS;
    float* Cp_v = small0 + 1 * bufS;
    float* Bp_a = small0 + 2 * bufS;
    float* Cp_a = small0 + 3 * bufS;
    float* pool_base = small0 + 4 * bufS;
    const size_t bufP = (size_t)BATCH * 64 * DCH;        // 98,304 floats
    float* pool_v = pool_base;
    float* pool_a = pool_base + bufP;
    const size_t bufY = (size_t)BATCH * 64 * OUTC;       // 262,144 floats
    float* yvb = pool_base + 2 * bufP;
    float* yab = yvb + bufY;
    const size_t bufC = (size_t)BATCH * NCH * DCH * NST; // 786,432 floats
    float* Ebuf = yab + bufY;
    float* Pbuf = Ebuf + bufC;
    float* Sbuf = Pbuf + bufC;
    float* y_v   = xvn;     // reuse
    float* y_a   = xan;     // reuse
    float* out_v = dt_v;    // reuse
    float* out_a = dt_a;    // reuse

    // 1) LayerNorm both modalities
    ln_kernel<<<SEQ / 4, 128, 0, stream>>>(xv, ln_v_g, ln_v_b, xvn);
    ln_kernel<<<SEQ / 4, 128, 0, stream>>>(xa, ln_a_g, ln_a_b, xan);

    // 2) u / dt GEMMs (WMMA fp32; wave = 16-row strip x 96 cols; B in LDS)
    const int gB = (SEQ / 16) / 8;                       // 128 blocks of 8 waves
    const size_t shm6 = (size_t)DCH * 96 * sizeof(float);     // 36 KB
    gemm_wmma_f32<6,0,0><<<gB, 256, shm6, stream>>>(xvn, DCH, Win_v, DCH, 1, nullptr, u_v,  DCH, SEQ, DCH);
    gemm_wmma_f32<6,1,0><<<gB, 256, shm6, stream>>>(xvn, DCH, Wdt_v, DCH, 1, nullptr, dt_v, DCH, SEQ, DCH);
    gemm_wmma_f32<6,0,0><<<gB, 256, shm6, stream>>>(xan, DCH, Win_a, DCH, 1, nullptr, u_a,  DCH, SEQ, DCH);
    gemm_wmma_f32<6,1,0><<<gB, 256, shm6, stream>>>(xan, DCH, Wdt_a, DCH, 1, nullptr, dt_a, DCH, SEQ, DCH);

    // 2b) cross B/C projections (N=8): v's SSM uses a's projections, and vice versa
    proj8_kernel<<<SEQ * NST / 256, 256, 0, stream>>>(xan, WB_v, Bp_v);
    proj8_kernel<<<SEQ * NST / 256, 256, 0, stream>>>(xan, WC_v, Cp_v);
    proj8_kernel<<<SEQ * NST / 256, 256, 0, stream>>>(xvn, WB_a, Bp_a);
    proj8_kernel<<<SEQ * NST / 256, 256, 0, stream>>>(xvn, WC_a, Cp_a);

    // 3) chunk-parallel scans (64 chunks of 16): v then a, y overwrites xvn/xan
    scan_pass1<<<BATCH * NCH, DCH, 0, stream>>>(dt_v, u_v, Bp_v, Alog_v, Ebuf, Pbuf);
    scan_pass2<<<BATCH * DCH * NST / 256, 256, 0, stream>>>(Ebuf, Pbuf, Sbuf);
    scan_pass3<<<BATCH * NCH, DCH, 0, stream>>>(dt_v, u_v, Bp_v, Cp_v, Alog_v, Dskip_v, Sbuf, y_v);
    scan_pass1<<<BATCH * NCH, DCH, 0, stream>>>(dt_a, u_a, Bp_a, Alog_a, Ebuf, Pbuf);
    scan_pass2<<<BATCH * DCH * NST / 256, 256, 0, stream>>>(Ebuf, Pbuf, Sbuf);
    scan_pass3<<<BATCH * NCH, DCH, 0, stream>>>(dt_a, u_a, Bp_a, Cp_a, Alog_a, Dskip_a, Sbuf, y_a);

    // 4) output projections with BCHW residual (WMMA)
    gemm_wmma_f32<6,0,1><<<gB, 256, shm6, stream>>>(y_v, DCH, Wout_v, DCH, 1, xv, out_v, DCH, SEQ, DCH);
    gemm_wmma_f32<6,0,1><<<gB, 256, shm6, stream>>>(y_a, DCH, Wout_a, DCH, 1, xa, out_a, DCH, SEQ, DCH);

    // 5) fused avg/max pooling to (b, p=64, c=96)
    pool_kernel<<<BATCH * 64 * DCH / 256, 256, 0, stream>>>(out_v, out_a, pool_v, pool_a);

    // 6) separable cost-volume conv (WMMA): yv = pool_v @ Wv^T, ya = pool_a @ Wa^T,
    //    each in two 128-column halves. B(k,n) = conv_w[(n+n0)*192 + koff + k].
    const int gC = (BATCH * 64 / 16) / 8;                // 8 blocks
    const size_t shm8 = (size_t)DCH * 128 * sizeof(float);    // 48 KB
    gemm_wmma_f32<8,0,0><<<gC, 256, shm8, stream>>>(pool_v, DCH, conv_w,                 1, 2 * DCH, nullptr, yvb,       OUTC, BATCH * 64, DCH);
    gemm_wmma_f32<8,0,0><<<gC, 256, shm8, stream>>>(pool_v, DCH, conv_w + 128 * 2 * DCH, 1, 2 * DCH, nullptr, yvb + 128, OUTC, BATCH * 64, DCH);
    gemm_wmma_f32<8,0,0><<<gC, 256, shm8, stream>>>(pool_a, DCH, conv_w + DCH,           1, 2 * DCH, nullptr, yab,       OUTC, BATCH * 64, DCH);
    gemm_wmma_f32<8,0,0><<<gC, 256, shm8, stream>>>(pool_a, DCH, conv_w + 128 * 2 * DCH + DCH, 1, 2 * DCH, nullptr, yab + 128, OUTC, BATCH * 64, DCH);

    // 7) separable bilinear upsample + outer-sum + bias -> 256 MB NT streaming write
    final_kernel<<<BATCH * OUTC, 256, 0, stream>>>(yvb, yab, conv_b, out);
}